// PseudoGridND_21234318312200
// MI455X (gfx1250) — compile-verified
//
#include <hip/hip_runtime.h>

// 3-D multilinear (trilinear) interpolated table lookup, MI455X (gfx1250).
//
// Roofline: table (64 MB) is L2-resident (192 MB L2); stream traffic
// (unList 24 MB in, out 64 MB out) is marked non-temporal so it cannot
// evict the table. HBM floor ~152 MB -> ~6.5 us @ 23.3 TB/s. Pure gather
// workload: no shared WMMA operand exists (corner matrix is per-point,
// data-dependent), so the optimal mix is b128 vmem gathers + NT streams.

typedef float v4f __attribute__((ext_vector_type(4)));

#define TDIM  128
#define TCH   8      // channels; one table row = 8 floats = 32 bytes = 2x v4f

__global__ __launch_bounds__(256)
void trilerp_grid_kernel(const float* __restrict__ un,
                         const float* __restrict__ table,
                         float* __restrict__ out,
                         int batch)
{
    int b = blockIdx.x * blockDim.x + threadIdx.x;
    if (b >= batch) return;

    // ---- streamed coordinate read (read-once): non-temporal loads ----
    const float* up = un + 3ull * (unsigned)b;
    float ux = __builtin_nontemporal_load(up + 0);
    float uy = __builtin_nontemporal_load(up + 1);
    float uz = __builtin_nontemporal_load(up + 2);

    // scaled = clip(u,0,1) * (dim-1); i0 = clip(floor(scaled), 0, dim-2);
    // frac = scaled - i0   (matches reference exactly, frac may reach 1.0)
    float sx = fminf(fmaxf(ux, 0.0f), 1.0f) * 127.0f;
    float sy = fminf(fmaxf(uy, 0.0f), 1.0f) * 127.0f;
    float sz = fminf(fmaxf(uz, 0.0f), 1.0f) * 127.0f;

    float fx0 = fminf(floorf(sx), 126.0f);
    float fy0 = fminf(floorf(sy), 126.0f);
    float fz0 = fminf(floorf(sz), 126.0f);

    int ix = (int)fx0, iy = (int)fy0, iz = (int)fz0;
    float fx = sx - fx0, fy = sy - fy0, fz = sz - fz0;
    float gx = 1.0f - fx, gy = 1.0f - fy, gz = 1.0f - fz;

    // corner weights: w(bx,by,bz) = (bx?fx:gx)*(by?fy:gy)*(bz?fz:gz)
    float wyz00 = gy * gz, wyz01 = gy * fz, wyz10 = fy * gz, wyz11 = fy * fz;
    float w000 = gx * wyz00, w001 = gx * wyz01, w010 = gx * wyz10, w011 = gx * wyz11;
    float w100 = fx * wyz00, w101 = fx * wyz01, w110 = fx * wyz10, w111 = fx * wyz11;

    // flat row index (units of 8-float channel rows); z-stride 1, y-stride 128,
    // x-stride 16384. Corner pairs (bz=0,1) are 64 contiguous bytes.
    int r000 = (ix * TDIM + iy) * TDIM + iz;
    int r010 = r000 + TDIM;
    int r100 = r000 + TDIM * TDIM;
    int r110 = r100 + TDIM;

    const v4f* __restrict__ tab = (const v4f*)table; // row r -> tab[2r], tab[2r+1]

    // ---- issue all 16 independent b128 gathers up front (max MLP) ----
    // These stay regular-temporal so the 64 MB table remains L2-resident.
    v4f c000a = tab[2 * (size_t)r000 + 0], c000b = tab[2 * (size_t)r000 + 1];
    v4f c001a = tab[2 * (size_t)r000 + 2], c001b = tab[2 * (size_t)r000 + 3];
    v4f c010a = tab[2 * (size_t)r010 + 0], c010b = tab[2 * (size_t)r010 + 1];
    v4f c011a = tab[2 * (size_t)r010 + 2], c011b = tab[2 * (size_t)r010 + 3];
    v4f c100a = tab[2 * (size_t)r100 + 0], c100b = tab[2 * (size_t)r100 + 1];
    v4f c101a = tab[2 * (size_t)r100 + 2], c101b = tab[2 * (size_t)r100 + 3];
    v4f c110a = tab[2 * (size_t)r110 + 0], c110b = tab[2 * (size_t)r110 + 1];
    v4f c111a = tab[2 * (size_t)r110 + 2], c111b = tab[2 * (size_t)r110 + 3];

    // ---- weighted accumulation (vector FMAs, scalar splat) ----
    v4f a0 = w000 * c000a;            v4f a1 = w000 * c000b;
    a0 += w001 * c001a;               a1 += w001 * c001b;
    a0 += w010 * c010a;               a1 += w010 * c010b;
    a0 += w011 * c011a;               a1 += w011 * c011b;
    a0 += w100 * c100a;               a1 += w100 * c100b;
    a0 += w101 * c101a;               a1 += w101 * c101b;
    a0 += w110 * c110a;               a1 += w110 * c110b;
    a0 += w111 * c111a;               a1 += w111 * c111b;

    // ---- streamed output (write-once): non-temporal b128 stores ----
    v4f* op = (v4f*)(out + 8ull * (unsigned)b);
    __builtin_nontemporal_store(a0, op + 0);
    __builtin_nontemporal_store(a1, op + 1);
}

extern "C" void kernel_launch(void* const* d_in, const int* in_sizes, int n_in,
                              void* d_out, int out_size, void* d_ws, size_t ws_size,
                              hipStream_t stream) {
    const float* unList = (const float*)d_in[0];   // [B, 3] f32
    const float* table  = (const float*)d_in[1];   // [128,128,128,8] f32
    float* out          = (float*)d_out;           // [B, 8] f32

    int batch = in_sizes[0] / 3;
    int threads = 256;
    int blocks = (batch + threads - 1) / threads;
    trilerp_grid_kernel<<<blocks, threads, 0, stream>>>(unList, table, out, batch);
}